// HyperConvLayer_23510650978832
// MI455X (gfx1250) — compile-verified
//
#include <hip/hip_runtime.h>
#include <stdint.h>

#define N_NODES 50000
#define N_EDGES 640000
#define DIM     128
#define LDA     132   // padded LDS stride for the A tile (conflict-free b64 reads)

typedef __attribute__((ext_vector_type(2))) float v2f;
typedef __attribute__((ext_vector_type(8))) float v8f;

// ---------------------------------------------------------------------------
// Kernel 1: zero summed[N*128] and cnt[N] in workspace (total divisible by 4)
// ---------------------------------------------------------------------------
__global__ void zero_ws_kernel(float4* __restrict__ ws, int n4) {
    int i = blockIdx.x * blockDim.x + threadIdx.x;
    if (i < n4) ws[i] = make_float4(0.f, 0.f, 0.f, 0.f);
}

// ---------------------------------------------------------------------------
// Kernel 2: one wave32 per edge. Lane l adds x[col][4l..4l+3] into
// summed[row][4l..4l+3] via native f32 global atomics; lane 0 bumps cnt[row].
// ---------------------------------------------------------------------------
__global__ __launch_bounds__(256)
void scatter_edges_kernel(const float* __restrict__ x,
                          const long long* __restrict__ ei,   // int64 [2,E]
                          float* __restrict__ summed,
                          float* __restrict__ cnt) {
    int gtid = blockIdx.x * blockDim.x + threadIdx.x;
    int e    = gtid >> 5;
    int lane = gtid & 31;
    if (e >= N_EDGES) return;

    long long row = ei[e];                       // destination
    long long col = ei[(long long)N_EDGES + e];  // source

    const float4* src = (const float4*)(x + col * DIM);
    float4 v = src[lane];                        // global_load_b128

    float* dst = summed + row * DIM + lane * 4;
    unsafeAtomicAdd(dst + 0, v.x);               // global_atomic_add_f32
    unsafeAtomicAdd(dst + 1, v.y);
    unsafeAtomicAdd(dst + 2, v.z);
    unsafeAtomicAdd(dst + 3, v.w);
    if (lane == 0) unsafeAtomicAdd(cnt + row, 1.0f);
}

// ---------------------------------------------------------------------------
// Kernel 3: out = relu((summed/max(cnt,1)) @ W + b) with V_WMMA_F32_16X16X4_F32.
// Block = 8 waves; block b, wave w computes the 16x16 tile (rows 16b.., cols 16w..).
// A tile (16x128) staged in LDS (stride 132); the mean division is folded into
// the A fragment as a per-row scale. B read from global (W is cache-resident).
// ---------------------------------------------------------------------------
__global__ __launch_bounds__(256)
void gemm_wmma_kernel(const float* __restrict__ summed,
                      const float* __restrict__ cnt,
                      const float* __restrict__ W,
                      const float* __restrict__ bias,
                      float* __restrict__ out) {
    __shared__ float lA[16 * LDA];

    const int tid    = threadIdx.x;
    const int base_m = blockIdx.x * 16;

    // Stage 16 rows x 128 cols of summed into LDS (512 float4s / 256 threads).
    for (int c = tid; c < (16 * DIM) / 4; c += 256) {
        int m  = c >> 5;            // 32 float4 per row
        int f4 = c & 31;
        float4 v = ((const float4*)(summed + (base_m + m) * DIM))[f4];
        float* dst = lA + m * LDA + f4 * 4;
        dst[0] = v.x; dst[1] = v.y; dst[2] = v.z; dst[3] = v.w;
    }
    __syncthreads();

    const int wave     = tid >> 5;        // tile_n index (0..7)
    const int lane     = tid & 31;
    const int m_local  = lane & 15;       // A-layout: both halves hold M=0..15
    const int kpair    = (lane >> 4) * 2; // lanes16-31 hold K=+2,+3
    const int n_local  = lane & 15;
    const int n_global = wave * 16 + n_local;

    const float c0   = cnt[base_m + m_local];
    const float invc = 1.0f / fmaxf(c0, 1.0f);

    v8f acc = {};
    #pragma unroll
    for (int kk = 0; kk < DIM; kk += 4) {
        // A fragment: v2f = {A[m][kk+kpair], A[m][kk+kpair+1]} * invc  (ds_load_b64)
        v2f a = *(const v2f*)(lA + m_local * LDA + kk + kpair);
        a.x *= invc;
        a.y *= invc;
        // B fragment: {W[kk+kpair][n], W[kk+kpair+1][n]}
        v2f bm;
        bm.x = W[(kk + kpair)     * DIM + n_global];
        bm.y = W[(kk + kpair + 1) * DIM + n_global];
        acc = __builtin_amdgcn_wmma_f32_16x16x4_f32(
            /*neg_a=*/false, a, /*neg_b=*/false, bm,
            /*c_mod=*/(short)0, acc, /*reuse_a=*/false, /*reuse_b=*/false);
    }

    const float bn       = bias[n_global];
    const int   row_base = (lane >> 4) * 8;   // C layout: VGPR r -> M=r (lo), M=r+8 (hi)
    #pragma unroll
    for (int r = 0; r < 8; ++r) {
        int   m = base_m + row_base + r;
        float v = acc[r] + bn;
        out[m * DIM + n_global] = fmaxf(v, 0.0f);
    }
}

// ---------------------------------------------------------------------------
extern "C" void kernel_launch(void* const* d_in, const int* in_sizes, int n_in,
                              void* d_out, int out_size, void* d_ws, size_t ws_size,
                              hipStream_t stream) {
    const float*     x    = (const float*)d_in[0];
    const long long* ei   = (const long long*)d_in[1];   // int64 edge_index [2,E]
    const float*     W    = (const float*)d_in[2];
    const float*     bias = (const float*)d_in[3];
    float*           out  = (float*)d_out;

    float* summed = (float*)d_ws;                 // N*128 floats
    float* cnt    = summed + (size_t)N_NODES * DIM;

    // 1) zero workspace
    int n4 = (N_NODES * DIM + N_NODES) / 4;       // 1,612,500 float4
    zero_ws_kernel<<<(n4 + 255) / 256, 256, 0, stream>>>((float4*)d_ws, n4);

    // 2) edge scatter (1 wave per edge -> 80000 blocks of 256)
    int scatter_blocks = (N_EDGES * 32) / 256;
    scatter_edges_kernel<<<scatter_blocks, 256, 0, stream>>>(x, ei, summed, cnt);

    // 3) WMMA GEMM + bias + relu (N = 3125 * 16 exactly)
    gemm_wmma_kernel<<<N_NODES / 16, 256, 0, stream>>>(summed, cnt, W, bias, out);
}